// Down_14156212208354
// MI455X (gfx1250) — compile-verified
//
#include <hip/hip_runtime.h>

// ---------------- problem constants (from reference) ----------------
constexpr int BN_   = 4;        // batch
constexpr int CIN_  = 32;
constexpr int COUT_ = 64;
constexpr int V_    = 40962;
constexpr int V_PAD = 40976;    // V rounded up to multiple of 16 (N-stride of feat)
constexpr int VIN_  = 163842;
constexpr int FQ_   = 81920;    // F = 2V-4
constexpr float EPS_ = 1e-5f;

typedef __attribute__((ext_vector_type(2))) float v2f;
typedef __attribute__((ext_vector_type(8))) float v8f;

// --------------------------------------------------------------------
// Kernel 1: fused G-spmm + EW/NS contraction.
// grad_ew[b,c,f] = sum_d EW[f,d] * sum_j Gval[(d*F+f)*3+j] * h[b,c,Gcol[...]]
// --------------------------------------------------------------------
__global__ void __launch_bounds__(256)
grad_kernel(const float* __restrict__ h, int rowStride, int C,
            const int* __restrict__ Gcol, const float* __restrict__ Gval,
            const float* __restrict__ EW, const float* __restrict__ NS,
            float* __restrict__ gew, float* __restrict__ gns)
{
    size_t idx   = (size_t)blockIdx.x * blockDim.x + threadIdx.x;
    size_t total = (size_t)BN_ * C * FQ_;
    if (idx >= total) return;
    int    f  = (int)(idx % FQ_);
    size_t bc = idx / FQ_;                       // b*C + c
    const float* hrow = h + bc * (size_t)rowStride;
    float ew = 0.f, ns = 0.f;
#pragma unroll
    for (int d = 0; d < 3; ++d) {
        size_t r = (size_t)d * FQ_ + f;
        float s = 0.f;
#pragma unroll
        for (int j = 0; j < 3; ++j) {
            int   col = Gcol[r * 3 + j];
            float w   = Gval[r * 3 + j];
            s += w * hrow[col];
        }
        ew += s * EW[f * 3 + d];
        ns += s * NS[f * 3 + d];
    }
    gew[idx] = ew;
    gns[idx] = ns;
}

// --------------------------------------------------------------------
// Kernel 2: assemble WMMA B operand feat[b][(c*4+k)][v] (N-stride V_PAD):
//   k=0 identity, k=1 laplacian (L-spmm), k=2/3 gv_ew/gv_ns (F-spmm).
// --------------------------------------------------------------------
__global__ void __launch_bounds__(256)
feat_kernel(const float* __restrict__ h, int rowStride, int C,
            const int* __restrict__ Lcol, const float* __restrict__ Lval,
            const int* __restrict__ Fcol, const float* __restrict__ Fval,
            const float* __restrict__ gew, const float* __restrict__ gns,
            float* __restrict__ feat)
{
    size_t idx   = (size_t)blockIdx.x * blockDim.x + threadIdx.x;
    size_t total = (size_t)BN_ * C * V_;
    if (idx >= total) return;
    int    v  = (int)(idx % V_);
    size_t bc = idx / V_;
    const float* hrow = h + bc * (size_t)rowStride;

    float ident = hrow[v];
    float lap = 0.f;
#pragma unroll
    for (int j = 0; j < 7; ++j) {
        int col = Lcol[(size_t)v * 7 + j];
        lap += Lval[(size_t)v * 7 + j] * hrow[col];
    }
    const float* ger = gew + bc * (size_t)FQ_;
    const float* gnr = gns + bc * (size_t)FQ_;
    float ge = 0.f, gn = 0.f;
#pragma unroll
    for (int j = 0; j < 6; ++j) {
        int   col = Fcol[(size_t)v * 6 + j];
        float w   = Fval[(size_t)v * 6 + j];
        ge += w * ger[col];
        gn += w * gnr[col];
    }
    float* frow = feat + (bc * 4) * (size_t)V_PAD + v;   // K index = c*4 + k
    frow[0]               = ident;
    frow[(size_t)V_PAD]   = lap;
    frow[(size_t)2*V_PAD] = ge;
    frow[(size_t)3*V_PAD] = gn;
}

// --------------------------------------------------------------------
// Kernel 3: out[b,o,v] = sum_K coeffs[o,K] * feat[b,K,v] + bias[o]
// via V_WMMA_F32_16X16X4_F32. Per wave: one 16-col tile, full 64 rows
// (4 accumulators). A staged in LDS. B loads are unconditional thanks
// to the V_PAD N-stride; pad columns of D are never stored (D column n
// depends only on B column n, so garbage stays in pad columns).
//   A lane layout (16x4 f32):  a[r] = A[m=lane&15][kk + 2*(lane>=16) + r]
//   B lane layout (4x16 f32):  b[r] = B[kk + 2*(lane>=16) + r][n=lane&15]
//   C/D layout: VGPR r, lane L -> D[m = r + 8*(L>=16)][n = L&15]
// --------------------------------------------------------------------
template<int K>
__global__ void __launch_bounds__(256)
gemm_wmma_kernel(const float* __restrict__ feat,     // [B][K][V_PAD]
                 const float* __restrict__ coeffs,   // [64][K]
                 const float* __restrict__ bias,     // [64]
                 float* __restrict__ out)            // [B][64][V]
{
    extern __shared__ float sA[];                    // 64*K floats
    for (int i = threadIdx.x; i < 64 * K; i += blockDim.x) sA[i] = coeffs[i];
    __syncthreads();

    const int lane = threadIdx.x & 31;
    const int wave = threadIdx.x >> 5;
    const int tilesPerBatch = V_PAD / 16;            // 2561
    const int tile = blockIdx.x * 8 + wave;
    const int b  = tile / tilesPerBatch;
    const int nt = tile - b * tilesPerBatch;
    if (b >= BN_) return;                            // wave-uniform exit

    const int nlane = lane & 15;
    const int halfk = (lane >> 4) << 1;              // 0 or 2
    const int n     = nt * 16 + nlane;
    const float* fb = feat + (size_t)b * K * V_PAD;

    v8f acc0 = {}, acc1 = {}, acc2 = {}, acc3 = {};

    for (int kk = 0; kk < K; kk += 4) {
        const int k0 = kk + halfk;
        // prefetch the B stream a couple of k-groups ahead (global_prefetch_b8)
        __builtin_prefetch(fb + (size_t)(k0 + 8) * V_PAD + n, 0, 1);
        v2f bv;
        bv.x = fb[(size_t)k0 * V_PAD + n];
        bv.y = fb[(size_t)(k0 + 1) * V_PAD + n];
        const float* ar = sA + (size_t)nlane * K + k0;
        v2f a0 = { ar[0],          ar[1] };
        v2f a1 = { ar[16 * K],     ar[16 * K + 1] };
        v2f a2 = { ar[32 * K],     ar[32 * K + 1] };
        v2f a3 = { ar[48 * K],     ar[48 * K + 1] };
        acc0 = __builtin_amdgcn_wmma_f32_16x16x4_f32(false, a0, false, bv, (short)0, acc0, false, false);
        acc1 = __builtin_amdgcn_wmma_f32_16x16x4_f32(false, a1, false, bv, (short)0, acc1, false, false);
        acc2 = __builtin_amdgcn_wmma_f32_16x16x4_f32(false, a2, false, bv, (short)0, acc2, false, false);
        acc3 = __builtin_amdgcn_wmma_f32_16x16x4_f32(false, a3, false, bv, (short)0, acc3, false, false);
    }

    if (n >= V_) return;                             // epilogue-only divergence
    float* ob = out + (size_t)b * 64 * V_ + n;
    const int mhalf = (lane >> 4) * 8;
#pragma unroll
    for (int r = 0; r < 8; ++r) {
        int m0 =      mhalf + r;  ob[(size_t)m0 * V_] = acc0[r] + bias[m0];
        int m1 = 16 + mhalf + r;  ob[(size_t)m1 * V_] = acc1[r] + bias[m1];
        int m2 = 32 + mhalf + r;  ob[(size_t)m2 * V_] = acc2[r] + bias[m2];
        int m3 = 48 + mhalf + r;  ob[(size_t)m3 * V_] = acc3[r] + bias[m3];
    }
}

// --------------------------------------------------------------------
// Kernel 4: per-channel mean / rstd over (batch, vertex).
// --------------------------------------------------------------------
__global__ void __launch_bounds__(256)
bn_stats_kernel(const float* __restrict__ y, float* __restrict__ stats)
{
    const int o = blockIdx.x;                        // channel
    __shared__ float ssum[256];
    __shared__ float ssq[256];
    float s = 0.f, q = 0.f;
    const size_t NV = (size_t)BN_ * V_;
    for (size_t i = threadIdx.x; i < NV; i += blockDim.x) {
        size_t b = i / V_, v = i - b * V_;
        float val = y[(b * 64 + o) * (size_t)V_ + v];
        s += val; q += val * val;
    }
    ssum[threadIdx.x] = s; ssq[threadIdx.x] = q;
    __syncthreads();
    for (int st = 128; st > 0; st >>= 1) {
        if ((int)threadIdx.x < st) {
            ssum[threadIdx.x] += ssum[threadIdx.x + st];
            ssq[threadIdx.x]  += ssq[threadIdx.x + st];
        }
        __syncthreads();
    }
    if (threadIdx.x == 0) {
        float ncnt = (float)NV;
        float mean = ssum[0] / ncnt;
        float var  = ssq[0] / ncnt - mean * mean;
        stats[o]      = mean;
        stats[64 + o] = rsqrtf(var + EPS_);
    }
}

// --------------------------------------------------------------------
// Kernel 5: normalize * gamma + beta, ReLU.
// --------------------------------------------------------------------
__global__ void __launch_bounds__(256)
bn_apply_kernel(const float* __restrict__ y, const float* __restrict__ stats,
                const float* __restrict__ gamma, const float* __restrict__ beta,
                float* __restrict__ out)
{
    size_t idx   = (size_t)blockIdx.x * blockDim.x + threadIdx.x;
    size_t total = (size_t)BN_ * 64 * V_;
    if (idx >= total) return;
    int o = (int)((idx / V_) % 64);
    float xn = (y[idx] - stats[o]) * stats[64 + o];
    float r  = xn * gamma[o] + beta[o];
    out[idx] = fmaxf(r, 0.f);
}

static inline int gridFor(size_t total, int block) {
    return (int)((total + block - 1) / block);
}

extern "C" void kernel_launch(void* const* d_in, const int* in_sizes, int n_in,
                              void* d_out, int out_size, void* d_ws, size_t ws_size,
                              hipStream_t stream)
{
    const float* x     = (const float*)d_in[0];
    const int*   Lcol  = (const int*)  d_in[2];
    const float* Lval  = (const float*)d_in[3];
    const int*   Gcol  = (const int*)  d_in[5];
    const float* Gval  = (const float*)d_in[6];
    const int*   Fcol  = (const int*)  d_in[8];
    const float* Fval  = (const float*)d_in[9];
    const float* EW    = (const float*)d_in[10];
    const float* NS    = (const float*)d_in[11];
    const float* co1   = (const float*)d_in[12];
    const float* bias1 = (const float*)d_in[13];
    const float* g1    = (const float*)d_in[14];
    const float* be1   = (const float*)d_in[15];
    const float* co2   = (const float*)d_in[16];
    const float* bias2 = (const float*)d_in[17];
    const float* g2    = (const float*)d_in[18];
    const float* be2   = (const float*)d_in[19];

    // ---- workspace layout (floats), sized for layer 2 (C=64) ----
    float* ws = (float*)d_ws;
    const size_t gradSz = (size_t)BN_ * COUT_ * FQ_;        // 20,971,520
    const size_t featSz = (size_t)BN_ * COUT_ * 4 * V_PAD;  // padded N-stride
    const size_t ySz    = (size_t)BN_ * COUT_ * V_;         // 10,486,272
    float* gew   = ws;
    float* gns   = gew  + gradSz;
    float* feat  = gns  + gradSz;
    float* ybuf  = feat + featSz;
    float* h2    = ybuf + ySz;
    float* stats = h2   + ySz;                              // 128 floats

    const int blk = 256;
    const int tilesPerBatch = V_PAD / 16;
    const int gemmBlocks = (BN_ * tilesPerBatch + 7) / 8;

    // ================= Layer 1 (C = 32, input = x, stride VIN) =================
    {
        const int C = CIN_;
        grad_kernel<<<gridFor((size_t)BN_ * C * FQ_, blk), blk, 0, stream>>>(
            x, VIN_, C, Gcol, Gval, EW, NS, gew, gns);
        feat_kernel<<<gridFor((size_t)BN_ * C * V_, blk), blk, 0, stream>>>(
            x, VIN_, C, Lcol, Lval, Fcol, Fval, gew, gns, feat);
        gemm_wmma_kernel<128><<<gemmBlocks, blk, 64 * 128 * sizeof(float), stream>>>(
            feat, co1, bias1, ybuf);
        bn_stats_kernel<<<64, blk, 0, stream>>>(ybuf, stats);
        bn_apply_kernel<<<gridFor(ySz, blk), blk, 0, stream>>>(
            ybuf, stats, g1, be1, h2);
    }

    // ================= Layer 2 (C = 64, input = h2, stride V) =================
    {
        const int C = COUT_;
        grad_kernel<<<gridFor((size_t)BN_ * C * FQ_, blk), blk, 0, stream>>>(
            h2, V_, C, Gcol, Gval, EW, NS, gew, gns);
        feat_kernel<<<gridFor((size_t)BN_ * C * V_, blk), blk, 0, stream>>>(
            h2, V_, C, Lcol, Lval, Fcol, Fval, gew, gns, feat);
        gemm_wmma_kernel<256><<<gemmBlocks, blk, 64 * 256 * sizeof(float), stream>>>(
            feat, co2, bias2, ybuf);
        bn_stats_kernel<<<64, blk, 0, stream>>>(ybuf, stats);
        bn_apply_kernel<<<gridFor(ySz, blk), blk, 0, stream>>>(
            ybuf, stats, g2, be2, (float*)d_out);
    }
}